// Quantizer_43576738186008
// MI455X (gfx1250) — compile-verified
//
#include <hip/hip_runtime.h>

typedef float     v2f  __attribute__((ext_vector_type(2)));
typedef float     v8f  __attribute__((ext_vector_type(8)));
typedef int       v4i_g __attribute__((vector_size(16)));          // GCC vector int4
typedef __attribute__((address_space(1))) v4i_g* as1_v4i_ptr;      // global int4*
typedef __attribute__((address_space(3))) v4i_g* as3_v4i_ptr;      // LDS int4*

#define ROWS   1024
#define COLS   1024
#define BS     16
#define NBLK   4096        // (1024/16)*(1024/16)
#define TSTEPS 32          // chunks per block (256/8)
#define NSTATE 256
#define CHUNK  8

#if __has_builtin(__builtin_amdgcn_global_load_async_to_lds_b128)
#define HAVE_ASYNC_LDS 1
#else
#define HAVE_ASYNC_LDS 0
#endif

__device__ __forceinline__ void wait_async_zero() {
#if __has_builtin(__builtin_amdgcn_s_wait_asynccnt)
  __builtin_amdgcn_s_wait_asynccnt(0);
#else
  asm volatile("s_wait_asynccnt 0x0" ::: "memory");
#endif
}

// One wave32 per 16x16 block: async-stage tile+codebook into LDS, WMMA cost
// GEMM -> LDS, Viterbi DP, backtrack, write reconstruction + state path.
__global__ __launch_bounds__(32) void viterbi_quant_kernel(
    const float* __restrict__ arr, const float* __restrict__ cbk,
    float* __restrict__ out)
{
  __shared__ __align__(16) float cost[TSTEPS * NSTATE];   // 32 KB
  __shared__ __align__(16) float tgt[TSTEPS * CHUNK];     // 1 KB (== block row-major)
  __shared__ __align__(16) float cbraw[NSTATE * CHUNK];   // 8 KB (codebook copy)
  __shared__ float cb2[NSTATE * CHUNK];                   // 8 KB (-2 * codebook)
  __shared__ float csq[NSTATE];                           // ||codebook[s]||^2
  __shared__ float dp[NSTATE];
  __shared__ float mval[16];
  __shared__ int   bp[TSTEPS * 16];                       // bp per (t, h=s>>4)
  __shared__ int   path[TSTEPS];

  const int l  = threadIdx.x;               // 0..31
  const int b  = blockIdx.x;                // 0..4095
  const int R0 = (b >> 6) * BS;
  const int C0 = (b & 63) * BS;
  const int lo = l & 15, hi = l >> 4;

  // ---- Stage inputs into LDS ------------------------------------------------
#if HAVE_ASYNC_LDS
  {
    // Tile: 16 rows x 64B, LDS layout tgt[row*16+col] == memory row-major.
    const int row = l >> 2, q = (l & 3) * 4;              // 16B per lane
    __builtin_amdgcn_global_load_async_to_lds_b128(
        (as1_v4i_ptr)(arr + (size_t)(R0 + row) * COLS + C0 + q),
        (as3_v4i_ptr)&tgt[row * BS + q], 0, 0);
    __builtin_amdgcn_global_load_async_to_lds_b128(
        (as1_v4i_ptr)(arr + (size_t)(R0 + row + 8) * COLS + C0 + q),
        (as3_v4i_ptr)&tgt[(row + 8) * BS + q], 0, 0);
    // Codebook: 8 KB = 16 issues of 512B.
    #pragma unroll
    for (int it = 0; it < 16; ++it) {
      const int off = it * 128 + l * 4;                   // float index
      __builtin_amdgcn_global_load_async_to_lds_b128(
          (as1_v4i_ptr)(cbk + off),
          (as3_v4i_ptr)&cbraw[off], 0, 0);
    }
    wait_async_zero();
  }
#else
  #pragma unroll
  for (int k = 0; k < 8; ++k) {
    int e = l + 32 * k;
    tgt[e] = arr[(R0 + (e >> 4)) * COLS + C0 + (e & 15)];
  }
  for (int i = l; i < NSTATE * CHUNK; i += 32) cbraw[i] = cbk[i];
#endif
  __syncthreads();

  for (int i = l; i < NSTATE * CHUNK; i += 32) cb2[i] = -2.0f * cbraw[i];
  #pragma unroll
  for (int k = 0; k < 8; ++k) {
    int s = l + 32 * k;
    float acc = 0.0f;
    #pragma unroll
    for (int c = 0; c < CHUNK; ++c) { float v = cbraw[s * CHUNK + c]; acc += v * v; }
    csq[s] = acc;
  }
  __syncthreads();

  // ---- Cost matrix via WMMA: cost[t][s] = csq[s] - 2*dot(tgt[t], cbk[s]) ----
  // (||target||^2 dropped: uniform shift per step, does not change argmins.)
  // unroll 4: keep accumulator VGPR pressure low -> more resident waves for
  // the latency-bound DP phase.
  #pragma unroll 4
  for (int nt = 0; nt < 16; ++nt) {
    const int s  = nt * 16 + lo;
    const float cs = csq[s];
    #pragma unroll
    for (int mt = 0; mt < 2; ++mt) {
      v8f c = {};
      // f32 K=4 WMMA, two steps cover K=8 exactly (matches f32 reference math).
      #pragma unroll
      for (int q = 0; q < 2; ++q) {
        const int kb = q * 4 + hi * 2;          // A 16x4 layout: lane-half -> K pair
        v2f a, bm;
        a.x  = tgt[(mt * 16 + lo) * CHUNK + kb];
        a.y  = tgt[(mt * 16 + lo) * CHUNK + kb + 1];
        bm.x = cb2[s * CHUNK + kb];             // B[k][s], mirrored layout
        bm.y = cb2[s * CHUNK + kb + 1];
        c = __builtin_amdgcn_wmma_f32_16x16x4_f32(false, a, false, bm,
                                                  (short)0, c, false, false);
      }
      // C/D layout: VGPR r -> row M = r + (lane/16)*8, col N = lane%16.
      #pragma unroll
      for (int r = 0; r < 8; ++r) {
        const int t = mt * 16 + hi * 8 + r;
        cost[t * NSTATE + s] = c[r] + cs;
      }
    }
  }
  __syncthreads();

  // ---- Viterbi DP -----------------------------------------------------------
  #pragma unroll
  for (int r = 0; r < 8; ++r) dp[l + 32 * r] = cost[l + 32 * r];
  __syncthreads();

  for (int t = 1; t < TSTEPS; ++t) {
    // Group-min over the 16 predecessors {p : p&15 == h}; strict-less keeps
    // the FIRST (smallest p) minimum, matching jnp.argmin.
    if (l < 16) {
      float best = dp[l];
      int   bi   = l;
      #pragma unroll
      for (int k = 1; k < 16; ++k) {
        float v = dp[l + 16 * k];
        if (v < best) { best = v; bi = l + 16 * k; }
      }
      mval[l]        = best;
      bp[t * 16 + l] = bi;                // bi == predecessor state itself
    }
    __syncthreads();
    float nv[8];
    #pragma unroll
    for (int r = 0; r < 8; ++r) {
      const int s = l + 32 * r;
      nv[r] = mval[s >> 4] + cost[t * NSTATE + s];
    }
    #pragma unroll
    for (int r = 0; r < 8; ++r) dp[l + 32 * r] = nv[r];
    __syncthreads();
  }

  // ---- Final argmin over 256 states (tie-break: smallest state) -------------
  float best = dp[l];
  int   bi   = l;
  #pragma unroll
  for (int r = 1; r < 8; ++r) {
    float v = dp[l + 32 * r];
    if (v < best) { best = v; bi = l + 32 * r; }
  }
  #pragma unroll
  for (int i = 0; i < 5; ++i) {
    const int off = 16 >> i;
    float ov = __shfl_xor(best, off, 32);
    int   oi = __shfl_xor(bi,   off, 32);
    if (ov < best || (ov == best && oi < bi)) { best = ov; bi = oi; }
  }

  // ---- Backtrack (lane 0) ---------------------------------------------------
  if (l == 0) {
    int s = bi;
    path[TSTEPS - 1] = s;
    for (int t = TSTEPS - 1; t >= 1; --t) {
      s = bp[t * 16 + (s >> 4)];
      path[t - 1] = s;
    }
  }
  __syncthreads();

  // ---- Outputs --------------------------------------------------------------
  float* recon = out;
  float* stout = out + ROWS * COLS;
  #pragma unroll
  for (int k = 0; k < 8; ++k) {
    const int e = l + 32 * k;
    const int t = e >> 3, c = e & 7;
    recon[(R0 + (e >> 4)) * COLS + C0 + (e & 15)] = cbraw[path[t] * CHUNK + c];
  }
  stout[b * TSTEPS + l] = (float)path[l];   // states, (br, bc, 32) flattened
}

extern "C" void kernel_launch(void* const* d_in, const int* in_sizes, int n_in,
                              void* d_out, int out_size, void* d_ws, size_t ws_size,
                              hipStream_t stream) {
  (void)in_sizes; (void)n_in; (void)d_ws; (void)ws_size; (void)out_size;
  const float* arr = (const float*)d_in[0];
  const float* cbk = (const float*)d_in[1];
  float* out = (float*)d_out;
  hipLaunchKernelGGL(viterbi_quant_kernel, dim3(NBLK), dim3(32), 0, stream,
                     arr, cbk, out);
}